// DFTQNN_55250459296416
// MI455X (gfx1250) — compile-verified
//
#include <hip/hip_runtime.h>
#include <hip/hip_bf16.h>

#define N_WIRES 24
#define DIM (1 << N_WIRES)
#define NB 1024
#define NT 256

typedef __attribute__((ext_vector_type(2))) float v2f;
typedef __attribute__((ext_vector_type(8))) float v8f;

// Per-Pauli-string parameters (gates and measurements share the layout).
// Gates:        c = cos(t/2), (wre,wim) = -i*sin(t/2)*i^{nY}
// Measurements: c unused,     (wre,wim) = i^{nY}
struct PP {
  int   xmask, zmask, hsh, pad0;
  float c, wre, wim, pad1;
};

// ---------------------------------------------------------------------------
// Wave32 reduction via V_WMMA_F32_16X16X4_F32.
// A: lane puts value v in A-vgpr0, 0 in A-vgpr1  -> sum(all A entries) = sum_lanes(v)
// B: all ones (4x16)                              -> D[i][j] = rowsum_i(A)
// Lane's 8 D vgprs cover rows 0-7 (lanes 0-15) or rows 8-15 (lanes 16-31),
// so per-lane sum of D regs + a SWAPX16 lane swizzle == full wave sum.
// ---------------------------------------------------------------------------
__device__ __forceinline__ float wave_reduce_wmma(float v) {
  v2f a; a[0] = v;    a[1] = 0.0f;
  v2f b; b[0] = 1.0f; b[1] = 1.0f;
  v8f c = {0.f, 0.f, 0.f, 0.f, 0.f, 0.f, 0.f, 0.f};
  v8f d = __builtin_amdgcn_wmma_f32_16x16x4_f32(
      /*neg_a=*/false, a, /*neg_b=*/false, b,
      /*c_mod=*/(short)0, c, /*reuse_a=*/false, /*reuse_b=*/false);
  float s = ((d[0] + d[1]) + (d[2] + d[3])) + ((d[4] + d[5]) + (d[6] + d[7]));
  // ds_swizzle SWAPX16: xor_mask=0x10, and_mask=0x1f  -> offset 0x401F
  float o = __int_as_float(__builtin_amdgcn_ds_swizzle(__float_as_int(s), 0x401F));
  return s + o;
}

// Deterministic 256-thread block reduction (8 waves). Result valid in thread 0.
__device__ __forceinline__ float block_reduce(float v, float* lds) {
  float w = wave_reduce_wmma(v);
  int lane = threadIdx.x & 31;
  int wid  = threadIdx.x >> 5;
  if (lane == 0) lds[wid] = w;
  __syncthreads();
  float tot = 0.0f;
  if (threadIdx.x == 0) {
#pragma unroll
    for (int i = 0; i < 8; ++i) tot += lds[i];
  }
  return tot;
}

__device__ __forceinline__ float2 cmix(float c, float wr, float wi,
                                       float2 a, float2 b, float s) {
  // c*a + (s*w)*b   (complex)
  float r0 = s * wr, r1 = s * wi;
  float2 r;
  r.x = c * a.x + r0 * b.x - r1 * b.y;
  r.y = c * a.y + r0 * b.y + r1 * b.x;
  return r;
}

// ---------------------------------------------------------------------------
// Precompute masks/phases for 10 gates + 3 measurements (13 tiny items).
// PennyLane convention: wire 0 = MSB -> bit = 1 << (23 - w).
// ---------------------------------------------------------------------------
__global__ void qnn_precompute(const float* __restrict__ theta,
                               const int*   __restrict__ ggen,
                               const int*   __restrict__ mcodes,
                               PP* __restrict__ gp, PP* __restrict__ mp) {
  int i = threadIdx.x;
  if (i >= 13) return;
  bool isGate = (i < 10);
  const int* codes = isGate ? (ggen + i * N_WIRES) : (mcodes + (i - 10) * N_WIRES);
  int xm = 0, zm = 0, ny = 0;
  for (int w = 0; w < N_WIRES; ++w) {
    int cde = codes[w];
    int bit = 1 << (N_WIRES - 1 - w);
    if (cde == 1 || cde == 2) xm |= bit;
    if (cde == 3 || cde == 2) zm |= bit;
    if (cde == 2) ++ny;
  }
  PP p;
  p.xmask = xm;
  p.zmask = zm;
  p.hsh   = xm ? (31 - __clz(xm)) : 0;
  p.pad0  = 0; p.pad1 = 0.f;
  // phase = i^{ny}
  const float prt[4] = {1.f, 0.f, -1.f, 0.f};
  const float pit[4] = {0.f, 1.f, 0.f, -1.f};
  float pr = prt[ny & 3], pi = pit[ny & 3];
  if (isGate) {
    float t  = theta[i];          // theta shape (10,1)
    float ch = cosf(0.5f * t);
    float sh = sinf(0.5f * t);
    p.c   = ch;
    p.wre = sh * pi;              // w = -i * sin(t/2) * (pr + i*pi)
    p.wim = -sh * pr;
    gp[i] = p;
  } else {
    p.c   = 0.f;
    p.wre = pr;
    p.wim = pi;
    mp[i - 10] = p;
  }
}

// ---------------------------------------------------------------------------
// Stage 1 of ||feature||^2: b128 loads, per-block partials (deterministic).
// ---------------------------------------------------------------------------
__global__ void qnn_norm_partial(const float4* __restrict__ f4, float* __restrict__ part) {
  __shared__ float lds[8];
  int t = blockIdx.x * blockDim.x + threadIdx.x;
  int stride = gridDim.x * blockDim.x;
  float acc = 0.f;
  for (int q = t; q < (DIM >> 2); q += stride) {
    float4 v = f4[q];
    acc = fmaf(v.x, v.x, acc);
    acc = fmaf(v.y, v.y, acc);
    acc = fmaf(v.z, v.z, acc);
    acc = fmaf(v.w, v.w, acc);
  }
  float tot = block_reduce(acc, lds);
  if (threadIdx.x == 0) part[blockIdx.x] = tot;
}

__global__ void qnn_norm_final(const float* __restrict__ part, float* __restrict__ invn) {
  __shared__ float lds[8];
  float acc = 0.f;
  for (int i = threadIdx.x; i < NB; i += blockDim.x) acc += part[i];
  float tot = block_reduce(acc, lds);
  if (threadIdx.x == 0) invn[0] = 1.0f / sqrtf(tot);
}

// psi cell q = psi4[q] holds amplitudes 2q (x,y) and 2q+1 (z,w).
__global__ void qnn_init_psi(const float2* __restrict__ f2,
                             const float* __restrict__ invn,
                             float4* __restrict__ psi4) {
  float s = invn[0];
  int t = blockIdx.x * blockDim.x + threadIdx.x;
  int stride = gridDim.x * blockDim.x;
  for (int q = t; q < (DIM >> 1); q += stride) {
    float2 f = f2[q];
    float4 r; r.x = f.x * s; r.y = 0.f; r.z = f.y * s; r.w = 0.f;
    psi4[q] = r;
  }
}

// ---------------------------------------------------------------------------
// One Pauli-rotation gate, in place, all accesses 16B (b128).
// psi'[j] = c*psi[j] + w*sign_j*psi[k],  sign_j = (-1)^popc(k & zmask), k=j^xmask
// psi'[k] = c*psi[k] + w*sign_k*psi[j],  sign_k = (-1)^popc(j & zmask)
// ---------------------------------------------------------------------------
__global__ void qnn_gate(float4* __restrict__ psi4, const PP* __restrict__ gp, int g) {
  PP p = gp[g];   // uniform -> scalar loads
  int t = blockIdx.x * blockDim.x + threadIdx.x;
  int stride = gridDim.x * blockDim.x;

  if (p.xmask == 0) {
    // Diagonal: psi'[j] = (c + w*sign_j)*psi[j]; 2 amplitudes per cell.
    for (int q = t; q < (DIM >> 1); q += stride) {
      float4 v = psi4[q];
      int j0 = q << 1;
      float s0 = (__popc(j0 & p.zmask) & 1) ? -1.f : 1.f;
      float s1 = (__popc((j0 | 1) & p.zmask) & 1) ? -1.f : 1.f;
      float cr0 = fmaf(p.wre, s0, p.c), ci0 = p.wim * s0;
      float cr1 = fmaf(p.wre, s1, p.c), ci1 = p.wim * s1;
      float4 r;
      r.x = cr0 * v.x - ci0 * v.y;
      r.y = cr0 * v.y + ci0 * v.x;
      r.z = cr1 * v.z - ci1 * v.w;
      r.w = cr1 * v.w + ci1 * v.z;
      psi4[q] = r;
    }
  } else if (p.hsh == 0) {
    // xmask == 1: pair (2q, 2q+1) lives in one cell.
    for (int q = t; q < (DIM >> 1); q += stride) {
      float4 v = psi4[q];
      int j = q << 1, k = j | 1;
      float2 a = {v.x, v.y}, b = {v.z, v.w};
      float sj = (__popc(k & p.zmask) & 1) ? -1.f : 1.f;
      float sk = (__popc(j & p.zmask) & 1) ? -1.f : 1.f;
      float2 na = cmix(p.c, p.wre, p.wim, a, b, sj);
      float2 nb = cmix(p.c, p.wre, p.wim, b, a, sk);
      float4 r; r.x = na.x; r.y = na.y; r.z = nb.x; r.w = nb.y;
      psi4[q] = r;
    }
  } else {
    // h >= 1: two adjacent pairs per thread; j0 even, j0+1 in same cell.
    int h  = p.hsh;
    int lo = (1 << h) - 1;
    bool swap = (p.xmask & 1) != 0;
    for (int u = t; u < (DIM >> 2); u += stride) {
      int i0 = u << 1;
      int j0 = ((i0 >> h) << (h + 1)) | (i0 & lo);   // even
      int k0 = j0 ^ p.xmask;
      int jq = j0 >> 1;
      int kq = k0 >> 1;
      float4 A = psi4[jq];
      float4 B = psi4[kq];
      float2 a0 = {A.x, A.y}, a1 = {A.z, A.w};
      float2 b0, b1;
      if (!swap) { b0.x = B.x; b0.y = B.y; b1.x = B.z; b1.y = B.w; }
      else       { b0.x = B.z; b0.y = B.w; b1.x = B.x; b1.y = B.y; }
      int j1 = j0 | 1, k1 = k0 ^ 1;
      float sj0 = (__popc(k0 & p.zmask) & 1) ? -1.f : 1.f;
      float sk0 = (__popc(j0 & p.zmask) & 1) ? -1.f : 1.f;
      float sj1 = (__popc(k1 & p.zmask) & 1) ? -1.f : 1.f;
      float sk1 = (__popc(j1 & p.zmask) & 1) ? -1.f : 1.f;
      float2 na0 = cmix(p.c, p.wre, p.wim, a0, b0, sj0);
      float2 nb0 = cmix(p.c, p.wre, p.wim, b0, a0, sk0);
      float2 na1 = cmix(p.c, p.wre, p.wim, a1, b1, sj1);
      float2 nb1 = cmix(p.c, p.wre, p.wim, b1, a1, sk1);
      float4 NA; NA.x = na0.x; NA.y = na0.y; NA.z = na1.x; NA.w = na1.y;
      float4 NBv;
      if (!swap) { NBv.x = nb0.x; NBv.y = nb0.y; NBv.z = nb1.x; NBv.w = nb1.y; }
      else       { NBv.x = nb1.x; NBv.y = nb1.y; NBv.z = nb0.x; NBv.w = nb0.y; }
      psi4[jq] = NA;
      psi4[kq] = NBv;
    }
  }
}

// ---------------------------------------------------------------------------
// <psi| M |psi> partials: blockIdx.y selects the measurement; b128 accesses.
// contrib = Re( conj(a) * (phase*sign) * b ),  b = psi[j ^ xmask]
// Generic cell path handles xmask==0 too (kq==jq, swap=0).
// ---------------------------------------------------------------------------
__global__ void qnn_meas_partial(const float4* __restrict__ psi4,
                                 const PP* __restrict__ mp,
                                 float* __restrict__ part) {
  __shared__ float lds[8];
  int m = blockIdx.y;
  PP p = mp[m];
  bool swap = (p.xmask & 1) != 0;
  int t = blockIdx.x * blockDim.x + threadIdx.x;
  int stride = gridDim.x * blockDim.x;
  float acc = 0.f;
  for (int q = t; q < (DIM >> 1); q += stride) {
    int j0 = q << 1;
    int k0 = j0 ^ p.xmask;
    float4 A = psi4[q];
    float4 B = psi4[k0 >> 1];
    float2 a0 = {A.x, A.y}, a1 = {A.z, A.w};
    float2 b0, b1;
    if (!swap) { b0.x = B.x; b0.y = B.y; b1.x = B.z; b1.y = B.w; }
    else       { b0.x = B.z; b0.y = B.w; b1.x = B.x; b1.y = B.y; }
    int k1 = k0 ^ 1;
    float s0 = (__popc(k0 & p.zmask) & 1) ? -1.f : 1.f;
    float s1 = (__popc(k1 & p.zmask) & 1) ? -1.f : 1.f;
    float wr0 = p.wre * s0, wi0 = p.wim * s0;
    float wr1 = p.wre * s1, wi1 = p.wim * s1;
    acc += a0.x * (wr0 * b0.x - wi0 * b0.y) + a0.y * (wr0 * b0.y + wi0 * b0.x);
    acc += a1.x * (wr1 * b1.x - wi1 * b1.y) + a1.y * (wr1 * b1.y + wi1 * b1.x);
  }
  float tot = block_reduce(acc, lds);
  if (threadIdx.x == 0) part[m * gridDim.x + blockIdx.x] = tot;
}

__global__ void qnn_meas_final(const float* __restrict__ part, float* __restrict__ out) {
  __shared__ float lds[8];
  int m = blockIdx.x;
  float acc = 0.f;
  for (int i = threadIdx.x; i < NB; i += blockDim.x) acc += part[m * NB + i];
  float tot = block_reduce(acc, lds);
  if (threadIdx.x == 0) out[m] = tot;
}

// ---------------------------------------------------------------------------
extern "C" void kernel_launch(void* const* d_in, const int* in_sizes, int n_in,
                              void* d_out, int out_size, void* d_ws, size_t ws_size,
                              hipStream_t stream) {
  const float* feature = (const float*)d_in[0];   // [DIM] f32
  const float* theta   = (const float*)d_in[1];   // [10,1] f32
  const int*   ggen    = (const int*)d_in[2];     // [10,24] i32
  const int*   mcodes  = (const int*)d_in[3];     // [3,24] i32
  float*       out     = (float*)d_out;           // [3] f32

  char* ws = (char*)d_ws;
  PP*     gp    = (PP*)ws;                        // 10 gate params
  PP*     mp    = gp + 10;                        // 3 measurement params
  float*  invn  = (float*)(ws + 512);             // 1 float
  float*  npart = (float*)(ws + 1024);            // NB floats
  float*  mpart = (float*)(ws + 8192);            // 3*NB floats
  float4* psi4  = (float4*)(ws + (size_t)(1u << 20)); // DIM/2 cells = 128 MB

  qnn_precompute<<<1, 32, 0, stream>>>(theta, ggen, mcodes, gp, mp);
  qnn_norm_partial<<<NB, NT, 0, stream>>>((const float4*)feature, npart);
  qnn_norm_final<<<1, NT, 0, stream>>>(npart, invn);
  qnn_init_psi<<<NB, NT, 0, stream>>>((const float2*)feature, invn, psi4);
  for (int g = 0; g < 10; ++g)
    qnn_gate<<<NB, NT, 0, stream>>>(psi4, gp, g);
  qnn_meas_partial<<<dim3(NB, 3), NT, 0, stream>>>(psi4, mp, mpart);
  qnn_meas_final<<<3, NT, 0, stream>>>(mpart, out);
}